// RegionProposalNetwork_22497038696818
// MI455X (gfx1250) — compile-verified
//
#include <hip/hip_runtime.h>
#include <hip/hip_bf16.h>

// ---------------------------------------------------------------------------
// RPN proposal filtering for MI455X (gfx1250, wave32).
//   B=32 images, 9 anchors * 100 * 100 = 90,000 candidates/image.
//   PRE_NMS = POST_NMS = 1000, NMS_THRESH = 0.7, MIN_SIZE = 1e-3.
// Kernel A: per-image radix top-k (sorted) -> decode/clip/sigmoid winners.
// Kernel B: per-image NMS (WMMA area-sum tiles + ballot bitmask in 128KB LDS,
//           async global->LDS staging) -> stable partition -> outputs.
// ---------------------------------------------------------------------------

#define NB      32
#define APER    9
#define FH      100
#define FW      100
#define HWSZ    (FH * FW)          // 10000
#define NANCH   (APER * HWSZ)      // 90000
#define KSEL    1000
#define KPAD    1024
#define EQCAP   256
#define BBOX_CLIP 4.135166556742356f   // log(1000/16)
#define NMS_T   0.7f
#define MIN_SZ  1e-3f

typedef __attribute__((ext_vector_type(2))) float v2f;
typedef __attribute__((ext_vector_type(8))) float v8f;

__device__ __forceinline__ unsigned key_of(float s) {
  unsigned u = __float_as_uint(s);
  return (u & 0x80000000u) ? ~u : (u | 0x80000000u);
}

// ===========================================================================
// Kernel A: top-1000 selection (sorted desc, index-asc tie-break) + decode.
// One 1024-thread workgroup per image.
// ===========================================================================
__global__ __launch_bounds__(1024) void rpn_topk_decode(
    const float* __restrict__ obj,      // [B, 9, 100, 100]
    const float* __restrict__ deltas,   // [B, 36, 100, 100]
    const float* __restrict__ anchors,  // [90000, 4]  (h,w,a order)
    const int* __restrict__ imh,
    const int* __restrict__ imw,
    float4* __restrict__ selb,          // [B, 1000]
    float* __restrict__ selp,           // [B, 1000]
    int* __restrict__ selv)             // [B, 1000]
{
  const int b   = blockIdx.x;
  const int tid = threadIdx.x;
  const float* ob = obj + (size_t)b * NANCH;

  __shared__ unsigned hist[256];
  __shared__ unsigned s_pref;
  __shared__ int      s_rem;
  __shared__ unsigned s_gt, s_eq, s_total;
  __shared__ unsigned long long skv[KPAD];
  __shared__ unsigned eqidx[EQCAP];

  // ---- 4-pass radix select: find the 1000th-largest key exactly ----------
  unsigned pref = 0;
  int remaining = KSEL;
  for (int pass = 0; pass < 4; ++pass) {
    const int shift = 24 - 8 * pass;
    for (int i = tid; i < 256; i += 1024) hist[i] = 0u;
    __syncthreads();
    for (int e = tid; e < NANCH; e += 1024) {          // source-linear: coalesced
      unsigned key = key_of(ob[e]);
      bool in = (pass == 0) || ((key >> (shift + 8)) == pref);
      if (in) atomicAdd(&hist[(key >> shift) & 255u], 1u);
    }
    __syncthreads();
    if (tid == 0) {
      unsigned cum = 0; int bin = 255;
      for (; bin >= 0; --bin) {
        unsigned c = hist[bin];
        if (cum + c >= (unsigned)remaining) break;
        cum += c;
      }
      if (bin < 0) bin = 0;
      s_pref = (pref << 8) | (unsigned)bin;
      s_rem  = remaining - (int)cum;
    }
    __syncthreads();
    pref = s_pref; remaining = s_rem;
    __syncthreads();
  }
  const unsigned T = pref;            // exact threshold key
  if (tid == 0) { s_gt = 0u; s_eq = 0u; }
  __syncthreads();

  // ---- compaction: strictly-greater elements + ties-at-threshold ---------
  for (int e = tid; e < NANCH; e += 1024) {
    unsigned key = key_of(ob[e]);
    if (key < T) continue;
    int a = e / HWSZ, hw = e % HWSZ;
    unsigned idx = (unsigned)(hw * APER + a);          // flattened (h,w,a) order
    if (key > T) {
      unsigned p = atomicAdd(&s_gt, 1u);
      if (p < (unsigned)KPAD)
        skv[p] = ((unsigned long long)key << 32) | (unsigned long long)(0xFFFFFFFFu - idx);
    } else {
      unsigned q = atomicAdd(&s_eq, 1u);
      if (q < EQCAP) eqidx[q] = idx;
    }
  }
  __syncthreads();
  if (tid == 0) {
    unsigned ne = s_eq; if (ne > EQCAP) ne = EQCAP;
    for (unsigned i = 1; i < ne; ++i) {                // smallest indices first
      unsigned v = eqidx[i]; int j = (int)i - 1;
      while (j >= 0 && eqidx[j] > v) { eqidx[j + 1] = eqidx[j]; --j; }
      eqidx[j + 1] = v;
    }
    unsigned base = s_gt; if (base > (unsigned)KPAD) base = KPAD;
    unsigned take = (unsigned)remaining; if (take > ne) take = ne;
    for (unsigned t = 0; t < take && (base + t) < (unsigned)KPAD; ++t)
      skv[base + t] = ((unsigned long long)T << 32) |
                      (unsigned long long)(0xFFFFFFFFu - eqidx[t]);
    unsigned tot = base + take; if (tot > (unsigned)KPAD) tot = KPAD;
    s_total = tot;
  }
  __syncthreads();
  for (int t = tid; t < KPAD; t += 1024)
    if ((unsigned)t >= s_total) skv[t] = 0ull;
  __syncthreads();

  // ---- bitonic sort, descending (key desc, index asc via ~idx) -----------
  for (unsigned k = 2; k <= (unsigned)KPAD; k <<= 1) {
    for (unsigned j = k >> 1; j > 0; j >>= 1) {
      unsigned ixj = (unsigned)tid ^ j;
      if (ixj > (unsigned)tid) {
        unsigned long long va = skv[tid], vb = skv[ixj];
        bool desc = (((unsigned)tid & k) == 0u);
        if (desc ? (va < vb) : (va > vb)) { skv[tid] = vb; skv[ixj] = va; }
      }
      __syncthreads();
    }
  }

  // ---- decode + clip + sigmoid only for the 1000 winners ------------------
  if (tid < KSEL) {
    unsigned long long comp = skv[tid];
    unsigned key = (unsigned)(comp >> 32);
    unsigned idx = 0xFFFFFFFFu - (unsigned)(comp & 0xFFFFFFFFull);
    float4 outb = make_float4(0.f, 0.f, 0.f, 0.f);
    float prob = 0.f; int vflag = 0;
    if (key != 0u && idx < (unsigned)NANCH) {
      unsigned u = (key & 0x80000000u) ? (key ^ 0x80000000u) : ~key;
      float score = __uint_as_float(u);
      prob = 1.0f / (1.0f + expf(-score));
      int a = (int)(idx % APER), hw = (int)(idx / APER);
      const float* anc = anchors + (size_t)idx * 4;
      float ax1 = anc[0], ay1 = anc[1], ax2 = anc[2], ay2 = anc[3];
      float aw = ax2 - ax1, ah = ay2 - ay1;
      float acx = ax1 + 0.5f * aw, acy = ay1 + 0.5f * ah;
      const float* dp = deltas + (size_t)b * (4 * APER * HWSZ)
                               + (size_t)(a * 4) * HWSZ + hw;
      float dx = dp[0];
      float dy = dp[HWSZ];
      float dw = fminf(dp[2 * HWSZ], BBOX_CLIP);
      float dh = fminf(dp[3 * HWSZ], BBOX_CLIP);
      float pcx = dx * aw + acx, pcy = dy * ah + acy;
      float pw = expf(dw) * aw, ph = expf(dh) * ah;
      float W = (float)imw[0], H = (float)imh[0];
      float x1 = fminf(fmaxf(pcx - 0.5f * pw, 0.f), W);
      float y1 = fminf(fmaxf(pcy - 0.5f * ph, 0.f), H);
      float x2 = fminf(fmaxf(pcx + 0.5f * pw, 0.f), W);
      float y2 = fminf(fmaxf(pcy + 0.5f * ph, 0.f), H);
      outb = make_float4(x1, y1, x2, y2);
      vflag = (((x2 - x1) >= MIN_SZ) && ((y2 - y1) >= MIN_SZ)) ? 1 : 0;
    }
    selb[(size_t)b * KSEL + tid] = outb;
    selp[(size_t)b * KSEL + tid] = prob;
    selv[(size_t)b * KSEL + tid] = vflag;
  }
}

// ===========================================================================
// Kernel B: NMS + stable partition + output. One 256-thread WG per image.
// Dynamic LDS layout (bytes):
// ===========================================================================
#define SB_OFF     0                       // float4 sboxes[1024]   : 16384
#define SAREA_OFF  16384                   // float  sarea [1024]   :  4096
#define SPROB_OFF  20480                   // float  sprob [1024]   :  4096
#define SMASK_OFF  24576                   // u32    smask[1024*32] : 131072
#define SVALID_OFF 155648                  // u32    svalid[32]     :   128
#define SKEEP_OFF  155776                  // u32    skeepw[32]     :   128
#define SWPREF_OFF 155904                  // u32    swpref[33]     :   132
#define SMEM_TOTAL 156064

__global__ __launch_bounds__(256) void rpn_nms_out(
    const float4* __restrict__ selb,
    const float* __restrict__ selp,
    const int* __restrict__ selv,
    float* __restrict__ out)
{
  const int b    = blockIdx.x;
  const int tid  = threadIdx.x;
  const int lane = tid & 31;
  const int wave = tid >> 5;

  extern __shared__ unsigned char smem[];
  float4*   sboxes = (float4*)  (smem + SB_OFF);
  float*    sarea  = (float*)   (smem + SAREA_OFF);
  float*    sprob  = (float*)   (smem + SPROB_OFF);
  unsigned* smask  = (unsigned*)(smem + SMASK_OFF);
  unsigned* svalid = (unsigned*)(smem + SVALID_OFF);
  unsigned* skeepw = (unsigned*)(smem + SKEEP_OFF);
  unsigned* swpref = (unsigned*)(smem + SWPREF_OFF);

  for (int i = tid; i < KPAD * 32; i += 256) smask[i] = 0u;
  if (tid < 32) { svalid[tid] = 0u; skeepw[tid] = 0u; }
  for (int r = tid; r < KPAD; r += 256) {
    if (r < KSEL) {
      sprob[r] = selp[(size_t)b * KSEL + r];
      if (selv[(size_t)b * KSEL + r]) atomicOr(&svalid[r >> 5], 1u << (r & 31));
    } else {
      sprob[r]  = 0.f;
      sboxes[r] = make_float4(0.f, 0.f, 0.f, 0.f);
    }
  }
  // Async global->LDS staging of the 1000 candidate boxes (16KB), one B128
  // transfer per row, tracked by ASYNCcnt.
  for (int r = tid; r < KSEL; r += 256) {
    unsigned laddr = (unsigned)(unsigned long long)&sboxes[r];   // LDS offset = addr[31:0]
    const float4* g = selb + (size_t)b * KSEL + r;
    asm volatile("global_load_async_to_lds_b128 %0, %1, off"
                 :: "v"(laddr), "v"(g) : "memory");
  }
  asm volatile("s_wait_asynccnt 0" ::: "memory");
  __syncthreads();
  for (int r = tid; r < KPAD; r += 256) {
    float4 bx = sboxes[r];
    sarea[r] = (bx.z - bx.x) * (bx.w - bx.y);
  }
  __syncthreads();

  // ---- suppression bit matrix: 16x16 wave tiles, WMMA area-outer-sum -----
  // C[m][n] = area_i0+m + area_j0+n  via  A=[area_i,1,0,0], B=[1,area_j,0,0]^T
  int tcount = 0;
  for (int ti = 0; ti < KPAD / 16; ++ti) {
    for (int tj = ti; tj < KPAD / 16; ++tj) {
      if ((tcount++ & 7) != wave) continue;
      const int i0 = ti << 4, j0 = tj << 4;
      const int half = lane >> 4, ln = lane & 15;
      v2f Av, Bv;
      Av.x = half ? 0.f : sarea[i0 + ln];   // A: lanes 0-15 K=0,1 ; 16-31 K=2,3
      Av.y = half ? 0.f : 1.0f;
      Bv.x = half ? 0.f : 1.0f;             // B: lanes 0-15 K=0,1 ; 16-31 K=2,3
      Bv.y = half ? 0.f : sarea[j0 + ln];
      v8f Cz = {0.f, 0.f, 0.f, 0.f, 0.f, 0.f, 0.f, 0.f};
      v8f S = __builtin_amdgcn_wmma_f32_16x16x4_f32(
          false, Av, false, Bv, (short)0, Cz, false, false);

      const int jg = j0 + ln;               // lane's column
      const float4 bj = sboxes[jg];
      const int wj = tj >> 1;
      const unsigned sh = (unsigned)(tj & 1) * 16u;
#pragma unroll
      for (int r = 0; r < 8; ++r) {
        const int ig = i0 + r + (half << 3);   // row per C/D VGPR layout
        const float4 bi = sboxes[ig];
        float lx = fmaxf(bi.x, bj.x), ly = fmaxf(bi.y, bj.y);
        float rx = fminf(bi.z, bj.z), ry = fminf(bi.w, bj.w);
        float inter = fmaxf(rx - lx, 0.f) * fmaxf(ry - ly, 0.f);
        float denom = S[r] - inter + 1e-12f;   // area_i + area_j - inter + eps
        bool sup = (inter > NMS_T * denom) && (jg > ig);
        unsigned bits = __builtin_amdgcn_ballot_w32(sup);
        if (lane == 0) {
          atomicOr(&smask[(i0 + r)     * 32 + wj], (bits & 0xFFFFu) << sh);
          atomicOr(&smask[(i0 + r + 8) * 32 + wj], (bits >> 16)     << sh);
        }
      }
    }
  }
  __syncthreads();

  // ---- greedy keep pass: wave 0, lane L owns removed-word L ---------------
  if (tid < 32) {
    unsigned removed = 0u, keepw = 0u;
    for (int i = 0; i < KSEL; ++i) {
      const int wi = i >> 5, bi = i & 31;
      unsigned rw = __shfl(removed, wi, 32);
      unsigned vw = svalid[wi];
      bool keep = ((vw >> bi) & 1u) && !((rw >> bi) & 1u);
      if (keep) {
        removed |= smask[i * 32 + lane];
        if (wi == lane) keepw |= (1u << bi);
      }
    }
    skeepw[lane] = keepw;
  }
  __syncthreads();
  if (tid == 0) {
    unsigned acc = 0;
    for (int w = 0; w < 32; ++w) { swpref[w] = acc; acc += __popc(skeepw[w]); }
    swpref[32] = acc;
  }
  __syncthreads();

  // ---- stable partition (kept first, score order) + masked writes ---------
  float* out_boxes = out;
  float* out_prob  = out + (size_t)NB * KSEL * 4;
  float* out_mask  = out + (size_t)NB * KSEL * 5;
  const unsigned nkeep = swpref[32];
  for (int r = tid; r < KSEL; r += 256) {
    const int w = r >> 5, bp = r & 31;
    const unsigned kw = skeepw[w];
    const bool keep = (kw >> bp) & 1u;
    const unsigned kb = swpref[w] + __popc(kw & ((1u << bp) - 1u));
    const unsigned pos = keep ? kb : (nkeep + (unsigned)r - kb);
    float4 bx = keep ? sboxes[r] : make_float4(0.f, 0.f, 0.f, 0.f);
    size_t o4 = ((size_t)b * KSEL + pos) * 4;
    out_boxes[o4 + 0] = bx.x; out_boxes[o4 + 1] = bx.y;
    out_boxes[o4 + 2] = bx.z; out_boxes[o4 + 3] = bx.w;
    out_prob[(size_t)b * KSEL + pos] = keep ? sprob[r] : 0.f;
    out_mask[(size_t)b * KSEL + pos] = keep ? 1.f : 0.f;
  }
}

// ===========================================================================
extern "C" void kernel_launch(void* const* d_in, const int* in_sizes, int n_in,
                              void* d_out, int out_size, void* d_ws, size_t ws_size,
                              hipStream_t stream) {
  const float* obj     = (const float*)d_in[0];   // [32,9,100,100]
  const float* deltas  = (const float*)d_in[1];   // [32,36,100,100]
  const float* anchors = (const float*)d_in[2];   // [90000,4]
  const int*   imh     = (const int*)d_in[3];
  const int*   imw     = (const int*)d_in[4];

  // workspace: selb [B*K] float4 | selp [B*K] float | selv [B*K] int  (~768KB)
  float4* selb = (float4*)d_ws;
  float*  selp = (float*)((char*)d_ws + (size_t)NB * KSEL * sizeof(float4));
  int*    selv = (int*)((char*)selp + (size_t)NB * KSEL * sizeof(float));

  rpn_topk_decode<<<NB, 1024, 0, stream>>>(obj, deltas, anchors, imh, imw,
                                           selb, selp, selv);
  rpn_nms_out<<<NB, 256, SMEM_TOTAL, stream>>>(selb, selp, selv, (float*)d_out);
}